// EntropyBottleneck_42545946034679
// MI455X (gfx1250) — compile-verified
//
#include <hip/hip_runtime.h>
#include <math.h>

// ---------------------------------------------------------------------------
// EntropyBottleneck forward for MI455X (gfx1250, wave32).
//   x: (16,192,64,64) f32.  One block per (b,c) plane of 64*64=4096 elements.
//   - Per-channel derived weights (softplus(matrix), tanh(factor)) computed
//     once per block into LDS, then held in registers (removes ~45
//     transcendentals/element vs naive formulation).
//   - x streamed through LDS with double-buffered GLOBAL_LOAD_ASYNC_TO_LDS_B128
//     (ASYNCcnt pipeline; per-lane self-contained, no barriers needed).
//   - 3x3 stage matmuls stay as register FMAs (WMMA layout repack between
//     tanh-gated stages costs more than the 9 FMAs/elem it would replace).
//   - bits block-reduction finishes on the matrix pipe: ones-matrix
//     V_WMMA_F32_16X16X4_F32 row-sum trick (full f32 precision).
//   - output stored non-temporally (write-once stream).
//   - deterministic: fixed-order reductions, no float atomics.
// ---------------------------------------------------------------------------

#define CHANNELS 192
#define HWPLANE  4096          // 64*64 elements per (b,c) plane
#define BLOCK    256           // 8 waves of 32
#define CHUNK    1024          // floats staged per pipeline step (BLOCK * 4)
#define NCHUNK   4             // HWPLANE / CHUNK

typedef float v2f __attribute__((ext_vector_type(2)));
typedef float v4f __attribute__((ext_vector_type(4)));
typedef float v8f __attribute__((ext_vector_type(8)));

// LDS layout for derived per-channel weights
enum { W_MED = 0,
       W_H0 = 1,  W_B0 = 4,  W_T0 = 7,
       W_H1 = 10, W_B1 = 19, W_T1 = 22,
       W_H2 = 25, W_B2 = 34, W_T2 = 37,
       W_H3 = 40, W_B3 = 49, W_T3 = 52,
       W_H4 = 55, W_B4 = 58, W_TOT = 64 };

__device__ __forceinline__ float fast_tanh(float x) {
#if __has_builtin(__builtin_amdgcn_tanhf)
  return __builtin_amdgcn_tanhf(x);      // V_TANH_F32 (gfx1250)
#else
  return tanhf(x);
#endif
}

__device__ __forceinline__ float fast_rcp(float x) {
#if __has_builtin(__builtin_amdgcn_rcpf)
  return __builtin_amdgcn_rcpf(x);
#else
  return 1.0f / x;
#endif
}

// numerically stable softplus (weights only; runs once per block)
__device__ __forceinline__ float softplus_f(float x) {
  return fmaxf(x, 0.0f) + log1pf(__expf(-fabsf(x)));
}

__device__ __forceinline__ float sigm(float x) {
  return fast_rcp(1.0f + __expf(-x));
}

// Async global->LDS copy of 16 bytes per lane (CDNA5 ASYNCcnt path).
__device__ __forceinline__ void async_copy_b128(uint32_t lds_off,
                                                const float* __restrict__ base,
                                                uint32_t byte_off) {
  asm volatile("global_load_async_to_lds_b128 %0, %1, %2"
               :: "v"(lds_off), "v"(byte_off), "s"(base)
               : "memory");
}

__global__ __launch_bounds__(BLOCK) void eb_main(
    const float* __restrict__ x,
    const float* __restrict__ m0, const float* __restrict__ b0, const float* __restrict__ f0,
    const float* __restrict__ m1, const float* __restrict__ b1, const float* __restrict__ f1,
    const float* __restrict__ m2, const float* __restrict__ b2, const float* __restrict__ f2,
    const float* __restrict__ m3, const float* __restrict__ b3, const float* __restrict__ f3,
    const float* __restrict__ m4, const float* __restrict__ b4,
    const float* __restrict__ q,
    float* __restrict__ out, float* __restrict__ partial)
{
  __shared__ float w[W_TOT];
  __shared__ float xbuf[2][CHUNK];
  __shared__ float red[BLOCK];

  const int tid = threadIdx.x;
  const int p   = blockIdx.x;          // plane index: b*CHANNELS + c
  const int c   = p % CHANNELS;

  // ---- per-channel derived weights (once per block) -----------------------
  if (tid < 9) {
    w[W_H1 + tid] = softplus_f(m1[c * 9 + tid]);
    w[W_H2 + tid] = softplus_f(m2[c * 9 + tid]);
    w[W_H3 + tid] = softplus_f(m3[c * 9 + tid]);
  }
  if (tid < 3) {
    w[W_H0 + tid] = softplus_f(m0[c * 3 + tid]);
    w[W_B0 + tid] = b0[c * 3 + tid];
    w[W_T0 + tid] = fast_tanh(f0[c * 3 + tid]);
    w[W_B1 + tid] = b1[c * 3 + tid];
    w[W_T1 + tid] = fast_tanh(f1[c * 3 + tid]);
    w[W_B2 + tid] = b2[c * 3 + tid];
    w[W_T2 + tid] = fast_tanh(f2[c * 3 + tid]);
    w[W_B3 + tid] = b3[c * 3 + tid];
    w[W_T3 + tid] = fast_tanh(f3[c * 3 + tid]);
    w[W_H4 + tid] = softplus_f(m4[c * 3 + tid]);
  }
  if (tid == 0) { w[W_B4] = b4[c]; w[W_MED] = q[c * 3 + 1]; }

  // ---- kick off async staging of chunk 0 (independent of weights) ---------
  const float* xp = x + (size_t)p * HWPLANE;
  const uint32_t lds0 = (uint32_t)(uintptr_t)(&xbuf[0][tid * 4]);
  const uint32_t lds1 = (uint32_t)(uintptr_t)(&xbuf[1][tid * 4]);
  async_copy_b128(lds0, xp, (uint32_t)(tid * 16));

  __syncthreads();

  // ---- hoist weights into registers ---------------------------------------
  float H0[3], Bb0[3], T0[3], H1[9], Bb1[3], T1[3];
  float H2[9], Bb2[3], T2[3], H3[9], Bb3[3], T3[3], H4[3];
#pragma unroll
  for (int i = 0; i < 3; ++i) {
    H0[i] = w[W_H0 + i]; Bb0[i] = w[W_B0 + i]; T0[i] = w[W_T0 + i];
    Bb1[i] = w[W_B1 + i]; T1[i] = w[W_T1 + i];
    Bb2[i] = w[W_B2 + i]; T2[i] = w[W_T2 + i];
    Bb3[i] = w[W_B3 + i]; T3[i] = w[W_T3 + i];
    H4[i] = w[W_H4 + i];
  }
#pragma unroll
  for (int i = 0; i < 9; ++i) {
    H1[i] = w[W_H1 + i]; H2[i] = w[W_H2 + i]; H3[i] = w[W_H3 + i];
  }
  const float Bb4 = w[W_B4];
  const float med = w[W_MED];

  auto mlp = [&](float v) -> float {
    float a0 = fmaf(H0[0], v, Bb0[0]);
    float a1 = fmaf(H0[1], v, Bb0[1]);
    float a2 = fmaf(H0[2], v, Bb0[2]);
    a0 = fmaf(T0[0], fast_tanh(a0), a0);
    a1 = fmaf(T0[1], fast_tanh(a1), a1);
    a2 = fmaf(T0[2], fast_tanh(a2), a2);

    float c0 = fmaf(H1[0], a0, fmaf(H1[1], a1, fmaf(H1[2], a2, Bb1[0])));
    float c1 = fmaf(H1[3], a0, fmaf(H1[4], a1, fmaf(H1[5], a2, Bb1[1])));
    float c2 = fmaf(H1[6], a0, fmaf(H1[7], a1, fmaf(H1[8], a2, Bb1[2])));
    c0 = fmaf(T1[0], fast_tanh(c0), c0);
    c1 = fmaf(T1[1], fast_tanh(c1), c1);
    c2 = fmaf(T1[2], fast_tanh(c2), c2);

    float d0 = fmaf(H2[0], c0, fmaf(H2[1], c1, fmaf(H2[2], c2, Bb2[0])));
    float d1 = fmaf(H2[3], c0, fmaf(H2[4], c1, fmaf(H2[5], c2, Bb2[1])));
    float d2 = fmaf(H2[6], c0, fmaf(H2[7], c1, fmaf(H2[8], c2, Bb2[2])));
    d0 = fmaf(T2[0], fast_tanh(d0), d0);
    d1 = fmaf(T2[1], fast_tanh(d1), d1);
    d2 = fmaf(T2[2], fast_tanh(d2), d2);

    float e0 = fmaf(H3[0], d0, fmaf(H3[1], d1, fmaf(H3[2], d2, Bb3[0])));
    float e1 = fmaf(H3[3], d0, fmaf(H3[4], d1, fmaf(H3[5], d2, Bb3[1])));
    float e2 = fmaf(H3[6], d0, fmaf(H3[7], d1, fmaf(H3[8], d2, Bb3[2])));
    e0 = fmaf(T3[0], fast_tanh(e0), e0);
    e1 = fmaf(T3[1], fast_tanh(e1), e1);
    e2 = fmaf(T3[2], fast_tanh(e2), e2);

    return fmaf(H4[0], e0, fmaf(H4[1], e1, fmaf(H4[2], e2, Bb4)));
  };

  float* op = out + (size_t)p * HWPLANE;
  float bsum = 0.0f;

  // ---- double-buffered ASYNCcnt pipeline over 4 chunks --------------------
  // Each lane reads back exactly the LDS bytes its own wave's async load
  // wrote, so s_wait_asynccnt alone orders producer->consumer (no barrier).
#pragma unroll
  for (int j = 0; j < NCHUNK; ++j) {
    if (j + 1 < NCHUNK) {
      const uint32_t lb = ((j + 1) & 1) ? lds1 : lds0;
      async_copy_b128(lb, xp, (uint32_t)((j + 1) * CHUNK * 4 + tid * 16));
      asm volatile("s_wait_asynccnt 1" ::: "memory");   // chunk j landed
    } else {
      asm volatile("s_wait_asynccnt 0" ::: "memory");   // last chunk landed
    }

    const v4f v4 =
        *reinterpret_cast<const v4f*>((j & 1) ? &xbuf[1][tid * 4]
                                              : &xbuf[0][tid * 4]);
    float vv[4] = { v4.x, v4.y, v4.z, v4.w };
    float ov[4];
#pragma unroll
    for (int e = 0; e < 4; ++e) {
      const float o  = rintf(vv[e] - med) + med;   // round-half-even == jnp.round
      const float lo = mlp(o - 0.5f);
      const float up = mlp(o + 0.5f);
      const float t  = lo + up;
      const float s  = (t > 0.0f) ? -1.0f : ((t < 0.0f) ? 1.0f : 0.0f);
      float lik = fabsf(sigm(s * up) - sigm(s * lo));
      lik = fmaxf(lik, 1e-9f);
      bsum -= __log2f(lik);
      ov[e] = o;
    }
    v4f o4 = { ov[0], ov[1], ov[2], ov[3] };
    // write-once streaming output: non-temporal store policy
    __builtin_nontemporal_store(o4,
        reinterpret_cast<v4f*>(op + j * CHUNK + tid * 4));
  }

  // ---- deterministic block reduction of bits ------------------------------
  red[tid] = bsum;
  __syncthreads();

#if __has_builtin(__builtin_amdgcn_wmma_f32_16x16x4_f32)
  // Matrix-pipe reduction: wave 0 folds 256 partials -> 32 lane values, then
  // one V_WMMA_F32_16X16X4_F32 with B = ones computes D[m,n] = v_m + v_{m+16}
  // (A is 16x4 striped: lanes 0-15 supply K0, lanes 16-31 supply K2; K1/K3=0).
  // Summing the 8 accumulator VGPRs in-lane gives the row-group sums; one
  // shfl_xor(16) completes the 32-lane total.  Full f32, deterministic.
  if (tid < 32) {
    float s = 0.0f;
#pragma unroll
    for (int i = 0; i < 8; ++i) s += red[tid + i * 32];
    v2f a;  a.x = s;     a.y = 0.0f;    // A[m,K0]=v_m (lanes 0-15), A[m,K2]=v_{m+16}
    v2f bo; bo.x = 1.0f; bo.y = 1.0f;   // B = ones(4x16)
    v8f acc = {};
    acc = __builtin_amdgcn_wmma_f32_16x16x4_f32(
        false, a, false, bo, (short)0, acc, false, false);
    float t = acc[0] + acc[1] + acc[2] + acc[3] +
              acc[4] + acc[5] + acc[6] + acc[7];
    t += __shfl_xor(t, 16, 32);         // combine the two row halves
    if (tid == 0) partial[p] = t;
  }
#else
#pragma unroll
  for (int s = BLOCK / 2; s > 0; s >>= 1) {
    if (tid < s) red[tid] += red[tid + s];
    __syncthreads();
  }
  if (tid == 0) partial[p] = red[0];
#endif
}

__global__ __launch_bounds__(BLOCK) void eb_reduce(const float* __restrict__ partial,
                                                   int n, float* __restrict__ bits_out)
{
  __shared__ double red[BLOCK];
  double s = 0.0;
  for (int i = threadIdx.x; i < n; i += BLOCK) s += (double)partial[i];
  red[threadIdx.x] = s;
  __syncthreads();
#pragma unroll
  for (int k = BLOCK / 2; k > 0; k >>= 1) {
    if (threadIdx.x < k) red[threadIdx.x] += red[threadIdx.x + k];
    __syncthreads();
  }
  if (threadIdx.x == 0) bits_out[0] = (float)red[0];
}

extern "C" void kernel_launch(void* const* d_in, const int* in_sizes, int n_in,
                              void* d_out, int out_size, void* d_ws, size_t ws_size,
                              hipStream_t stream)
{
  // setup_inputs() dict order:
  const float* m0 = (const float*)d_in[0];
  const float* b0 = (const float*)d_in[1];
  const float* f0 = (const float*)d_in[2];
  const float* m1 = (const float*)d_in[3];
  const float* b1 = (const float*)d_in[4];
  const float* f1 = (const float*)d_in[5];
  const float* m2 = (const float*)d_in[6];
  const float* b2 = (const float*)d_in[7];
  const float* f2 = (const float*)d_in[8];
  const float* m3 = (const float*)d_in[9];
  const float* b3 = (const float*)d_in[10];
  const float* f3 = (const float*)d_in[11];
  const float* m4 = (const float*)d_in[12];
  const float* b4 = (const float*)d_in[13];
  const float* q  = (const float*)d_in[14];
  const float* x  = (const float*)d_in[15];

  float* out     = (float*)d_out;
  float* partial = (float*)d_ws;                 // nplanes floats of scratch

  const int nplanes = in_sizes[15] / HWPLANE;    // 16*192 = 3072

  eb_main<<<nplanes, BLOCK, 0, stream>>>(x,
      m0, b0, f0, m1, b1, f1, m2, b2, f2, m3, b3, f3, m4, b4, q,
      out, partial);

  // bits scalar is the last element of d_out (tuple: out flat, then bits)
  eb_reduce<<<1, BLOCK, 0, stream>>>(partial, nplanes,
                                     out + (size_t)(out_size - 1));
}